// CenterNetDecoder_58171037057665
// MI455X (gfx1250) — compile-verified
//
#include <hip/hip_runtime.h>
#include <hip/hip_bf16.h>
#include <cstdint>

#define B_ 32
#define C_ 80
#define H_ 160
#define W_ 160
#define HW_ (H_ * W_)
#define K_ 100
#define KPAD 112            /* 7 WMMA tiles of 16 rows */
#define NPLANES (B_ * C_)
#define STRIP 16
#define NSTRIP (H_ / STRIP)
#define SROWS (STRIP + 2)
#define CAP 6400
#define NTHREADS 256
#define MINSCORE 0.05f
#define STRIDE_F 4.0f
#define NEGBIG  (-1.0e30f)
#define REMOVED (-2.0e30f)

typedef unsigned int v4u __attribute__((ext_vector_type(4)));
typedef int          v4i __attribute__((ext_vector_type(4)));
typedef int          v8i __attribute__((ext_vector_type(8)));
typedef float        v2f __attribute__((ext_vector_type(2)));
typedef float        v8f __attribute__((ext_vector_type(8)));

#if defined(__has_builtin)
#if __has_builtin(__builtin_amdgcn_tensor_load_to_lds)
#define HAVE_TDM 1
#endif
#endif

// ---- CDNA5 async global->LDS copy (gfx1250) --------------------
__device__ __forceinline__ void async_load_b128(uint32_t lds_off, uint32_t goff,
                                                unsigned long long sbase) {
    asm volatile("global_load_async_to_lds_b128 %0, %1, %2"
                 :: "v"(lds_off), "v"(goff), "s"(sbase) : "memory");
}
__device__ __forceinline__ void wait_async() {
    asm volatile("s_wait_asynccnt 0" ::: "memory");
}

#if HAVE_TDM
// ---- CDNA5 Tensor Data Mover: one 2D-tile descriptor per strip ---------
__device__ __forceinline__ void tdm_load_strip(unsigned long long ga,
                                               uint32_t lds_bytes, int rows) {
    v4u g0;
    g0.x = 1u;                                            // count=1, user mode
    g0.y = lds_bytes;                                     // lds_addr
    g0.z = (uint32_t)(ga & 0xFFFFFFFFull);                // global_addr[31:0]
    g0.w = (uint32_t)((ga >> 32) & 0x01FFFFFFull)         // global_addr[56:32]
         | (2u << 30);                                    // type = 2 (image)
    v8i g1;
    g1[0] = (int)(2u << 16);                              // data_size = 4 bytes
    g1[1] = (int)(((unsigned)W_ & 0xFFFFu) << 16);        // tensor_dim0 = W_
    g1[2] = (int)(((unsigned)rows & 0xFFFFu) << 16);      // tensor_dim1 = rows
    g1[3] = (int)(((unsigned)W_ & 0xFFFFu) << 16);        // tile_dim0 = W_
    g1[4] = rows;                                         // tile_dim1 = rows
    g1[5] = W_;                                           // tensor_dim0_stride
    g1[6] = 0;
    g1[7] = 0;
    v4i z4 = {0, 0, 0, 0};                                // groups 2/3 unused (2D)
#if __clang_major__ >= 23
    v8i z8 = {0, 0, 0, 0, 0, 0, 0, 0};
    __builtin_amdgcn_tensor_load_to_lds(g0, g1, z4, z4, z8, 0);
#else
    __builtin_amdgcn_tensor_load_to_lds(g0, g1, z4, z4, 0);
#endif
}

__device__ __forceinline__ void tdm_issue_strip(unsigned long long hm_base,
                                                uint32_t lds_base_bytes, int strip) {
    const int y0 = strip * STRIP;
    const int start_row = (strip == 0) ? 0 : (y0 - 1);
    const int dst_row   = (strip == 0) ? 1 : 0;
    int rows = SROWS;
    if (strip == 0) rows -= 1;
    if (strip == NSTRIP - 1) rows -= 1;
    const unsigned long long ga =
        hm_base + (unsigned long long)(start_row * W_) * 4ull;
    tdm_load_strip(ga, lds_base_bytes + (uint32_t)(dst_row * W_ * 4), rows);
}
#endif  // HAVE_TDM

// =======================================================================
// Kernel A: per-(b,c) plane — TDM double-buffered 3x3 NMS + plane top-100
// =======================================================================
__global__ __launch_bounds__(NTHREADS)
void centernet_plane_topk(const float* __restrict__ heat,
                          float* __restrict__ ws_scores,
                          int* __restrict__ ws_idx) {
    __shared__ __align__(16) float s_strip[2][SROWS * W_]; // 23.0 KB (double buf)
    __shared__ float          s_cval[CAP];                 // 25.6 KB
    __shared__ unsigned short s_cidx[CAP];                 // 12.8 KB
    __shared__ float s_rval[NTHREADS / 32];
    __shared__ int   s_rslot[NTHREADS / 32];
    __shared__ int   s_rsidx[NTHREADS / 32];
    __shared__ int   s_cnt;

    const int tid = threadIdx.x;
    const int plane = blockIdx.x;
    const float* hm = heat + (size_t)plane * HW_;
    const unsigned long long hm_base = (unsigned long long)(uintptr_t)hm;

    if (tid == 0) s_cnt = 0;

#if HAVE_TDM
    const uint32_t base0 = (uint32_t)(uintptr_t)&s_strip[0][0];
    const uint32_t base1 = (uint32_t)(uintptr_t)&s_strip[1][0];
    if (tid < W_) s_strip[0][tid] = NEGBIG;     // top halo pad for strip 0
    if (tid < 32) tdm_issue_strip(hm_base, base0, 0);   // prime pipeline
#endif
    __syncthreads();

    for (int s = 0; s < NSTRIP; ++s) {
#if HAVE_TDM
        if (tid < 32) {
            if (s + 1 < NSTRIP) {
                tdm_issue_strip(hm_base, ((s + 1) & 1) ? base1 : base0, s + 1);
                __builtin_amdgcn_s_wait_tensorcnt(1);   // strip s landed (in-order)
            } else {
                __builtin_amdgcn_s_wait_tensorcnt(0);
            }
        }
        __syncthreads();
        if (s == NSTRIP - 1) {  // bottom halo pad of final strip's buffer
            if (tid < W_) s_strip[s & 1][(SROWS - 1) * W_ + tid] = NEGBIG;
            __syncthreads();
        }
        const float* sb = &s_strip[s & 1][0];
#else
        const int y0f = s * STRIP;
        for (int ch = tid; ch < SROWS * (W_ / 4); ch += NTHREADS) {
            const int br = ch / (W_ / 4);
            const int bc = ch % (W_ / 4);
            const int gy = y0f - 1 + br;
            if (gy >= 0 && gy < H_) {
                const uint32_t goff = (uint32_t)((gy * W_ + bc * 4) * sizeof(float));
                const uint32_t ldso =
                    (uint32_t)(uintptr_t)&s_strip[0][br * W_ + bc * 4];
                async_load_b128(ldso, goff, hm_base);
            } else {
                float* p = &s_strip[0][br * W_ + bc * 4];
                p[0] = NEGBIG; p[1] = NEGBIG; p[2] = NEGBIG; p[3] = NEGBIG;
            }
        }
        wait_async();
        __syncthreads();
        const float* sb = &s_strip[0][0];
#endif
        const int y0 = s * STRIP;
        if (s + 1 < NSTRIP)
            __builtin_prefetch(hm + (size_t)(y0 + 2 * STRIP) * W_, 0, 0);

        for (int p = tid; p < STRIP * W_; p += NTHREADS) {
            const int lr = p / W_;
            const int x  = p % W_;
            const float* rowm = &sb[(lr + 0) * W_];
            const float* row  = &sb[(lr + 1) * W_];
            const float* rowp = &sb[(lr + 2) * W_];
            const float c = row[x];
            float m = NEGBIG;
            if (x > 0) {
                m = fmaxf(m, rowm[x - 1]);
                m = fmaxf(m, row [x - 1]);
                m = fmaxf(m, rowp[x - 1]);
            }
            m = fmaxf(m, rowm[x]);
            m = fmaxf(m, rowp[x]);
            if (x < W_ - 1) {
                m = fmaxf(m, rowm[x + 1]);
                m = fmaxf(m, row [x + 1]);
                m = fmaxf(m, rowp[x + 1]);
            }
            if (c >= m) {
                const int slot = atomicAdd(&s_cnt, 1);
                if (slot < CAP) {
                    s_cval[slot] = c;
                    s_cidx[slot] = (unsigned short)((y0 + lr) * W_ + x);
                }
            }
        }
        __syncthreads();
    }

    int n = s_cnt; if (n > CAP) n = CAP;

    for (int k = 0; k < K_; ++k) {
        float bv = NEGBIG; int bslot = -1; int bsidx = 0x7fffffff;
        for (int i = tid; i < n; i += NTHREADS) {
            const float v = s_cval[i];
            const int  si = (int)s_cidx[i];
            if (v > bv || (v == bv && si < bsidx)) { bv = v; bslot = i; bsidx = si; }
        }
        for (int o = 16; o > 0; o >>= 1) {
            const float ov = __shfl_down(bv, o);
            const int   os = __shfl_down(bslot, o);
            const int  osi = __shfl_down(bsidx, o);
            if (ov > bv || (ov == bv && osi < bsidx)) { bv = ov; bslot = os; bsidx = osi; }
        }
        if ((tid & 31) == 0) {
            const int w = tid >> 5;
            s_rval[w] = bv; s_rslot[w] = bslot; s_rsidx[w] = bsidx;
        }
        __syncthreads();
        if (tid == 0) {
            bv = s_rval[0]; bslot = s_rslot[0]; bsidx = s_rsidx[0];
            for (int w = 1; w < NTHREADS / 32; ++w) {
                if (s_rval[w] > bv || (s_rval[w] == bv && s_rsidx[w] < bsidx)) {
                    bv = s_rval[w]; bslot = s_rslot[w]; bsidx = s_rsidx[w];
                }
            }
            float outv = 0.0f; int outi = 0;
            if (bslot >= 0) { outv = bv; outi = bsidx; s_cval[bslot] = REMOVED; }
            ws_scores[(size_t)plane * K_ + k] = outv;
            ws_idx  [(size_t)plane * K_ + k] = outi;
        }
        __syncthreads();
    }
}

// =======================================================================
// Kernel B: per-batch global top-100 + WMMA bbox decode
// bbox(det) = [xs ys w h] x Bmat, Bmat(4x16) = STRIDE * [[1,0,1,0],[0,1,0,1],
// [-.5,0,.5,0],[0,-.5,0,.5]] in cols 0..3 -> V_WMMA_F32_16X16X4_F32, 7 tiles.
// =======================================================================
__global__ __launch_bounds__(NTHREADS)
void centernet_final(const float* __restrict__ ws_scores,
                     const int* __restrict__ ws_idx,
                     const float* __restrict__ off,
                     const float* __restrict__ wh,
                     float* __restrict__ out) {
    __shared__ __align__(16) float s_val[C_ * K_];        // 32 KB
    __shared__ float s_rval[NTHREADS / 32];
    __shared__ int   s_rsel[NTHREADS / 32];
    __shared__ float s_xs[KPAD], s_ys[KPAD], s_w0[KPAD], s_w1[KPAD], s_dsc[KPAD];

    const int tid = threadIdx.x;
    const int b = blockIdx.x;
    const unsigned long long src_base =
        (unsigned long long)(uintptr_t)(ws_scores + (size_t)b * C_ * K_);

    for (int ch = tid; ch < (C_ * K_) / 4; ch += NTHREADS) {
        const uint32_t goff = (uint32_t)(ch * 16);
        const uint32_t ldso = (uint32_t)(uintptr_t)&s_val[ch * 4];
        async_load_b128(ldso, goff, src_base);
    }
    if (tid >= K_ && tid < KPAD) {   // zero the WMMA pad rows (100..111)
        s_xs[tid] = 0.0f; s_ys[tid] = 0.0f; s_w0[tid] = 0.0f; s_w1[tid] = 0.0f;
        s_dsc[tid] = 0.0f;
    }
    wait_async();
    __syncthreads();

    const float* offb = off + (size_t)b * 2 * HW_;
    const float* whb  = wh  + (size_t)b * 2 * HW_;
    float* out_sc = out;                    // [B,K]
    float* out_cl = out + B_ * K_;          // [B,K]
    float* out_bb = out + 2 * B_ * K_;      // [B,K,4]

    for (int k = 0; k < K_; ++k) {
        float bv = NEGBIG; int bsel = 0x7fffffff;
        for (int i = tid; i < C_ * K_; i += NTHREADS) {
            const float v = s_val[i];
            if (v > bv || (v == bv && i < bsel)) { bv = v; bsel = i; }
        }
        for (int o = 16; o > 0; o >>= 1) {
            const float ov = __shfl_down(bv, o);
            const int   os = __shfl_down(bsel, o);
            if (ov > bv || (ov == bv && os < bsel)) { bv = ov; bsel = os; }
        }
        if ((tid & 31) == 0) { s_rval[tid >> 5] = bv; s_rsel[tid >> 5] = bsel; }
        __syncthreads();
        if (tid == 0) {
            bv = s_rval[0]; bsel = s_rsel[0];
            for (int w = 1; w < NTHREADS / 32; ++w) {
                if (s_rval[w] > bv || (s_rval[w] == bv && s_rsel[w] < bsel)) {
                    bv = s_rval[w]; bsel = s_rsel[w];
                }
            }
            float score = 0.0f; int cls = 0; int spat = 0;
            if (bsel < C_ * K_) {
                score = bv;
                cls = bsel / K_;
                spat = ws_idx[(size_t)b * C_ * K_ + bsel];
                s_val[bsel] = REMOVED;
            }
            const float fy = (float)(spat / W_);
            const float fx = (float)(spat % W_);
            s_xs[k]  = fx + offb[spat];
            s_ys[k]  = fy + offb[HW_ + spat];
            s_w0[k]  = whb[spat];
            s_w1[k]  = whb[HW_ + spat];
            s_dsc[k] = score;
            const bool valid = score > MINSCORE;
            out_sc[b * K_ + k] = valid ? score : -1.0f;
            out_cl[b * K_ + k] = valid ? (float)cls : -1.0f;
        }
        __syncthreads();
    }

    // ---- WMMA bbox transform: 7 tiles of 16 detections, one per wave ----
    const int wv = tid >> 5;
    const int lane = tid & 31;
    if (wv < KPAD / 16) {
        const int row = lane & 15;
        const int d = wv * 16 + row;
        // A (16x4 f32): lanes 0-15 -> K=0,1 (xs,ys); lanes 16-31 -> K=2,3 (w,h)
        v2f a;
        if (lane < 16) { a.x = s_xs[d]; a.y = s_ys[d]; }
        else           { a.x = s_w0[d]; a.y = s_w1[d]; }
        // B (4x16 f32): VGPR0 lanes 0-15 = row K0, lanes 16-31 = row K2;
        //               VGPR1 lanes 0-15 = row K1, lanes 16-31 = row K3.
        v2f bm;
        const int col = lane & 15;
        if (lane < 16) {
            bm.x = (col == 0 || col == 2) ? STRIDE_F : 0.0f;            // K0
            bm.y = (col == 1 || col == 3) ? STRIDE_F : 0.0f;            // K1
        } else {
            bm.x = (col == 0) ? -0.5f * STRIDE_F
                 : (col == 2) ?  0.5f * STRIDE_F : 0.0f;                // K2
            bm.y = (col == 1) ? -0.5f * STRIDE_F
                 : (col == 3) ?  0.5f * STRIDE_F : 0.0f;                // K3
        }
        v8f c = {};
        c = __builtin_amdgcn_wmma_f32_16x16x4_f32(
                false, a, false, bm, (short)0, c, false, false);
        // D (16x16 f32): VGPR r: lanes 0-15 -> (M=r, N=lane); 16-31 -> (M=r+8)
        const int n = lane & 15;
        const int mbase = (lane < 16) ? 0 : 8;
        if (n < 4) {
            for (int r = 0; r < 8; ++r) {
                const int det = wv * 16 + mbase + r;
                if (det < K_) {
                    const bool valid = s_dsc[det] > MINSCORE;
                    out_bb[(size_t)(b * K_ + det) * 4 + n] = valid ? c[r] : 0.0f;
                }
            }
        }
    }
}

extern "C" void kernel_launch(void* const* d_in, const int* in_sizes, int n_in,
                              void* d_out, int out_size, void* d_ws, size_t ws_size,
                              hipStream_t stream) {
    const float* heat = (const float*)d_in[0];   // [32,80,160,160]
    const float* off  = (const float*)d_in[1];   // [32,2,160,160]
    const float* wh   = (const float*)d_in[2];   // [32,2,160,160]
    float* out = (float*)d_out;                  // scores(3200)++classes(3200)++bboxes(12800)

    float* ws_scores = (float*)d_ws;                                       // 2560*100 floats
    int*   ws_idx    = (int*)((char*)d_ws + (size_t)NPLANES * K_ * sizeof(float));

    centernet_plane_topk<<<NPLANES, NTHREADS, 0, stream>>>(heat, ws_scores, ws_idx);
    centernet_final<<<B_, NTHREADS, 0, stream>>>(ws_scores, ws_idx, off, wh, out);
}